// SimpleGCN_58445914964185
// MI455X (gfx1250) — compile-verified
//
#include <hip/hip_runtime.h>
#include <hip/hip_bf16.h>
#include <math.h>

typedef __attribute__((ext_vector_type(2))) float v2f;
typedef __attribute__((ext_vector_type(8))) float v8f;

#define C_DIM 64
#define BATCH 2

__device__ __forceinline__ void atomAddF32(float* p, float v) {
  // lowers to GLOBAL_ATOMIC_ADD_F32 (device-scope, relaxed, non-returning)
  __hip_atomic_fetch_add(p, v, __ATOMIC_RELAXED, __HIP_MEMORY_SCOPE_AGENT);
}

// ---------------- degree / normalization ----------------
__global__ __launch_bounds__(256) void gcn_deg_init(unsigned* deg, int N) {
  int i = blockIdx.x * blockDim.x + threadIdx.x;
  if (i < N) deg[i] = 1u;  // self-loop contributes 1 to every destination
}

__global__ __launch_bounds__(256) void gcn_deg_count(const int* __restrict__ dst,
                                                     unsigned* deg, int E) {
  int e = blockIdx.x * blockDim.x + threadIdx.x;
  if (e < E) atomicAdd(&deg[dst[e]], 1u);
}

__global__ __launch_bounds__(256) void gcn_dinv(float* dinv, int N) {
  int i = blockIdx.x * blockDim.x + threadIdx.x;
  if (i < N) {
    unsigned d = ((const unsigned*)dinv)[i];  // in-place reinterpret (deg >= 1 always)
    dinv[i] = rsqrtf((float)d);
  }
}

// ---------------- dense lift: Y[M,64] = X[M,64] @ W[64,64]^T ----------------
// One wave -> 16 rows x 64 cols using V_WMMA_F32_16X16X4_F32.
// A frag (16x4 f32): lane(0-15)=row m, halves pick K pairs {0,1}/{2,3} -> contiguous float2.
// B frag (4x16 f32): lane(0-15)=col n, same K striping; B[k][n] = W[n][k] (row-major W).
// C/D frag: VGPR r holds rows m0+r (lanes 0-15) and m0+r+8 (lanes 16-31), col = lane&15.
__global__ __launch_bounds__(256) void gcn_gemm_wmma(const float* __restrict__ X,
                                                     const float* __restrict__ W,
                                                     float* __restrict__ Y, int M) {
  int wave = blockIdx.x * (blockDim.x >> 5) + (threadIdx.x >> 5);
  int lane = threadIdx.x & 31;
  int m0 = wave * 16;
  if (m0 >= M) return;  // wave-uniform: EXEC stays all-ones for WMMA

  int half = lane >> 4;   // 0 or 1
  int ml   = lane & 15;   // row (A) / col (B/D) within the 16-wide tile

  const float* xr = X + (size_t)(m0 + ml) * C_DIM;
  const float* wr = W + (size_t)ml * C_DIM;   // row n=ml of W (cols of B, n-tile 0)

  v8f acc0 = {}, acc1 = {}, acc2 = {}, acc3 = {};
#pragma unroll
  for (int kk = 0; kk < C_DIM; kk += 4) {
    int k2 = kk + 2 * half;
    v2f a  = *(const v2f*)(xr + k2);
    v2f b0 = *(const v2f*)(wr + k2);
    v2f b1 = *(const v2f*)(wr + 16 * C_DIM + k2);
    v2f b2 = *(const v2f*)(wr + 32 * C_DIM + k2);
    v2f b3 = *(const v2f*)(wr + 48 * C_DIM + k2);
    acc0 = __builtin_amdgcn_wmma_f32_16x16x4_f32(false, a, false, b0, (short)0, acc0, false, false);
    acc1 = __builtin_amdgcn_wmma_f32_16x16x4_f32(false, a, false, b1, (short)0, acc1, false, false);
    acc2 = __builtin_amdgcn_wmma_f32_16x16x4_f32(false, a, false, b2, (short)0, acc2, false, false);
    acc3 = __builtin_amdgcn_wmma_f32_16x16x4_f32(false, a, false, b3, (short)0, acc3, false, false);
  }
#pragma unroll
  for (int r = 0; r < 8; ++r) {
    int row = m0 + r + 8 * half;
    float* yr = Y + (size_t)row * C_DIM + ml;
    yr[0]  = acc0[r];
    yr[16] = acc1[r];
    yr[32] = acc2[r];
    yr[48] = acc3[r];
  }
}

// ---------------- zero fill ----------------
__global__ __launch_bounds__(256) void gcn_zero4(float4* p, int n4) {
  int i = blockIdx.x * blockDim.x + threadIdx.x;
  if (i < n4) p[i] = make_float4(0.f, 0.f, 0.f, 0.f);
}

// ---------------- edge scatter: agg[b,dst] += t[b,src] * dinv[src]*dinv[dst] ----------------
// One wave per edge; 64 channels = 32 lanes x float2; src/dst/norm are wave-uniform (SGPR).
__global__ __launch_bounds__(256) void gcn_scatter(const float* __restrict__ t,
                                                   const int* __restrict__ src,
                                                   const int* __restrict__ dst,
                                                   const float* __restrict__ dinv,
                                                   float* __restrict__ agg,
                                                   int E, int N) {
  int e = blockIdx.x * (blockDim.x >> 5) + (threadIdx.x >> 5);
  if (e >= E) return;
  int lane = threadIdx.x & 31;
  int s = src[e], d = dst[e];
  float nrm = dinv[s] * dinv[d];
#pragma unroll
  for (int b = 0; b < BATCH; ++b) {
    const float2 v = ((const float2*)(t + ((size_t)b * N + s) * C_DIM))[lane];
    float* p = agg + ((size_t)b * N + d) * C_DIM + lane * 2;
    atomAddF32(p,     v.x * nrm);
    atomAddF32(p + 1, v.y * nrm);
  }
}

// ---------------- epilogue: h += dinv[n]^2 * t + bias ; optional exact GELU ----------------
__global__ __launch_bounds__(256) void gcn_epilogue(float* __restrict__ h,
                                                    const float* __restrict__ t,
                                                    const float* __restrict__ dinv,
                                                    const float* __restrict__ bias,
                                                    int N, int total, int do_gelu) {
  int i = blockIdx.x * blockDim.x + threadIdx.x;
  if (i >= total) return;
  int c = i & (C_DIM - 1);
  int node = (i >> 6) % N;       // i = ((b*N + n)*64 + c)
  float self = dinv[node];
  self *= self;                  // self-loop norm = dinv[n]^2
  float x = h[i] + t[i] * self + bias[c];
  if (do_gelu) x = 0.5f * x * (1.0f + erff(x * 0.70710678118654752f));
  h[i] = x;
}

extern "C" void kernel_launch(void* const* d_in, const int* in_sizes, int n_in,
                              void* d_out, int out_size, void* d_ws, size_t ws_size,
                              hipStream_t stream) {
  const float* x  = (const float*)d_in[0];
  const int*   ei = (const int*)d_in[1];
  const float* W1 = (const float*)d_in[2];
  const float* b1 = (const float*)d_in[3];
  const float* W2 = (const float*)d_in[4];
  const float* b2 = (const float*)d_in[5];

  const int E = in_sizes[1] / 2;
  const int N = in_sizes[0] / (BATCH * C_DIM);
  const int M = BATCH * N;                 // 100000 rows, divisible by 16
  const int total = M * C_DIM;             // 6.4M elements
  const int* src = ei;
  const int* dst = ei + E;

  // workspace: dinv[N] | T[M*64] | H[M*64]   (~52 MB)
  char* ws = (char*)d_ws;
  float* dinv = (float*)ws;
  size_t off = ((size_t)N * sizeof(float) + 255) & ~(size_t)255;
  float* T = (float*)(ws + off);
  off += (((size_t)total * sizeof(float)) + 255) & ~(size_t)255;
  float* H = (float*)(ws + off);
  float* OUT = (float*)d_out;

  const int thr = 256;
  const int wavesPerBlk = thr / 32;
  const int gemmBlocks    = (M / 16 + wavesPerBlk - 1) / wavesPerBlk;
  const int scatterBlocks = (E + wavesPerBlk - 1) / wavesPerBlk;
  const int elemBlocks    = (total + thr - 1) / thr;
  const int elem4Blocks   = (total / 4 + thr - 1) / thr;

  // degrees (incl. self loops) -> dinv = rsqrt(deg)
  gcn_deg_init <<<(N + thr - 1) / thr, thr, 0, stream>>>((unsigned*)dinv, N);
  gcn_deg_count<<<(E + thr - 1) / thr, thr, 0, stream>>>(dst, (unsigned*)dinv, E);
  gcn_dinv     <<<(N + thr - 1) / thr, thr, 0, stream>>>(dinv, N);

  // ---- layer 1 ----
  gcn_gemm_wmma<<<gemmBlocks, thr, 0, stream>>>(x, W1, T, M);
  gcn_zero4    <<<elem4Blocks, thr, 0, stream>>>((float4*)H, total / 4);
  gcn_scatter  <<<scatterBlocks, thr, 0, stream>>>(T, src, dst, dinv, H, E, N);
  gcn_epilogue <<<elemBlocks, thr, 0, stream>>>(H, T, dinv, b1, N, total, 1);

  // ---- layer 2 ----
  gcn_gemm_wmma<<<gemmBlocks, thr, 0, stream>>>(H, W2, T, M);
  gcn_zero4    <<<elem4Blocks, thr, 0, stream>>>((float4*)OUT, total / 4);
  gcn_scatter  <<<scatterBlocks, thr, 0, stream>>>(T, src, dst, dinv, OUT, E, N);
  gcn_epilogue <<<elemBlocks, thr, 0, stream>>>(OUT, T, dinv, b2, N, total, 0);
}